// GraphPooledConvolutionalNetwork_77747497992817
// MI455X (gfx1250) — compile-verified
//
#include <hip/hip_runtime.h>
#include <math.h>

typedef __attribute__((ext_vector_type(2))) float v2f;
typedef __attribute__((ext_vector_type(8))) float v8f;

#define EPS_BN 1e-5f
#define EPS_ENT 1e-15f

// ---------------- elementwise / utility ----------------

__global__ void fill_f32(float* p, float v, long n) {
  long i = (long)blockIdx.x * blockDim.x + threadIdx.x;
  long st = (long)gridDim.x * blockDim.x;
  for (; i < n; i += st) p[i] = v;
}

__global__ void deg_edges_kernel(const int* __restrict__ dst, float* deg, int E) {
  int e = blockIdx.x * blockDim.x + threadIdx.x;
  if (e < E) atomicAdd(&deg[dst[e]], 1.0f);
}

__global__ void dinv_kernel(float* degdinv, int N) {
  int i = blockIdx.x * blockDim.x + threadIdx.x;
  if (i < N) {
    float d = degdinv[i] + 1.0f;  // +1 self loop
    degdinv[i] = rsqrtf(fmaxf(d, 1e-12f));
  }
}

// out[n,c] = dinv[n]^2 * h[n,c] + bias[c]   (self-loop term + bias)
__global__ void selfloop_bias_kernel(const float* __restrict__ h, const float* __restrict__ dinv,
                                     const float* __restrict__ bias, float* out, long N, int C) {
  long total = N * (long)C;
  long st = (long)gridDim.x * blockDim.x;
  for (long idx = (long)blockIdx.x * blockDim.x + threadIdx.x; idx < total; idx += st) {
    int c = (int)(idx % C);
    long n = idx / C;
    float dv = dinv[n];
    out[idx] = dv * dv * h[idx] + bias[c];
  }
}

// one wave per edge: out[dst,c] += coef * h[src,c]; coef = dinv[s]*dinv[d] or 1
__global__ void edge_scatter_kernel(const int* __restrict__ src, const int* __restrict__ dst,
                                    const float* __restrict__ dinv,
                                    const float* __restrict__ h, float* out, int E, int C) {
  int wpb = blockDim.x >> 5;
  int e = blockIdx.x * wpb + (threadIdx.x >> 5);
  if (e >= E) return;
  int lane = threadIdx.x & 31;
  int s = src[e], d = dst[e];
  float coef = dinv ? dinv[s] * dinv[d] : 1.0f;
  const float* hs = h + (long)s * C;
  float* od = out + (long)d * C;
  for (int c = lane * 4; c < C; c += 128) {
    float4 hv = *(const float4*)(hs + c);
    atomicAdd(&od[c + 0], coef * hv.x);
    atomicAdd(&od[c + 1], coef * hv.y);
    atomicAdd(&od[c + 2], coef * hv.z);
    atomicAdd(&od[c + 3], coef * hv.w);
  }
}

// ---------------- WMMA GEMM (fp32, 16x16x4) ----------------
// C[M,Nn] = A @ B (TRANS_A=0, A:[M,Kc]) or A^T @ B (TRANS_A=1, A:[Kc,M]); B:[Kc,Nn]
// grid: (M/16, Nn/16, splitK); 1 wave (32 threads) per block.
// k-loop unrolled x4 with 2 accumulators to cluster loads and relax the WMMA RAW chain.

__device__ __forceinline__ v8f wmma4(v2f a, v2f b, v8f c) {
  return __builtin_amdgcn_wmma_f32_16x16x4_f32(false, a, false, b, (short)0, c, false, false);
}

template <bool TRANS_A>
__device__ __forceinline__ v2f ld_a(const float* __restrict__ A, int ka, int m, int M, int Kc) {
  v2f a;
  if (TRANS_A) {
    a.x = A[(long)ka * M + m];
    a.y = A[(long)(ka + 1) * M + m];
  } else {
    const float* p = A + (long)m * Kc + ka;
    a.x = p[0];
    a.y = p[1];
  }
  return a;
}

__device__ __forceinline__ v2f ld_b(const float* __restrict__ B, int ka, int n, int Nn) {
  v2f b;
  b.x = B[(long)ka * Nn + n];
  b.y = B[(long)(ka + 1) * Nn + n];
  return b;
}

template <bool TRANS_A, bool ATOMIC>
__global__ void gemm_wmma_kernel(const float* __restrict__ A, const float* __restrict__ B,
                                 float* Cm, int M, int Nn, int Kc) {
  int lane = threadIdx.x & 31;
  int half = lane >> 4;
  int l16 = lane & 15;
  int m = blockIdx.x * 16 + l16;
  int n = blockIdx.y * 16 + l16;
  int S = gridDim.z;
  int chunk = (((Kc + S - 1) / S) + 3) & ~3;
  int k0 = blockIdx.z * chunk;
  int k1 = k0 + chunk; if (k1 > Kc) k1 = Kc;
  v8f acc0 = {};
  v8f acc1 = {};
  int kk = k0;
  for (; kk + 16 <= k1; kk += 16) {
    int ka = kk + 2 * half;
    v2f a0 = ld_a<TRANS_A>(A, ka +  0, m, M, Kc);
    v2f a1 = ld_a<TRANS_A>(A, ka +  4, m, M, Kc);
    v2f a2 = ld_a<TRANS_A>(A, ka +  8, m, M, Kc);
    v2f a3 = ld_a<TRANS_A>(A, ka + 12, m, M, Kc);
    v2f b0 = ld_b(B, ka +  0, n, Nn);
    v2f b1 = ld_b(B, ka +  4, n, Nn);
    v2f b2 = ld_b(B, ka +  8, n, Nn);
    v2f b3 = ld_b(B, ka + 12, n, Nn);
    acc0 = wmma4(a0, b0, acc0);
    acc1 = wmma4(a1, b1, acc1);
    acc0 = wmma4(a2, b2, acc0);
    acc1 = wmma4(a3, b3, acc1);
  }
  for (; kk < k1; kk += 4) {
    int ka = kk + 2 * half;
    v2f a = ld_a<TRANS_A>(A, ka, m, M, Kc);
    v2f b = ld_b(B, ka, n, Nn);
    acc0 = wmma4(a, b, acc0);
  }
#pragma unroll
  for (int v = 0; v < 8; ++v) {
    int mo = blockIdx.x * 16 + v + 8 * half;
    float r = acc0[v] + acc1[v];
    if (ATOMIC) atomicAdd(&Cm[(long)mo * Nn + n], r);
    else        Cm[(long)mo * Nn + n] = r;
  }
}

// ---------------- batchnorm ----------------

__global__ void bn_stats_kernel(const float* __restrict__ x, float* stats, long N, int C) {
  __shared__ float ssum[256];
  __shared__ float ssq[256];
  for (int i = threadIdx.x; i < C; i += blockDim.x) { ssum[i] = 0.f; ssq[i] = 0.f; }
  __syncthreads();
  long total = N * (long)C;
  long st = (long)gridDim.x * blockDim.x;
  for (long idx = (long)blockIdx.x * blockDim.x + threadIdx.x; idx < total; idx += st) {
    float v = x[idx];
    int c = (int)(idx % C);
    atomicAdd(&ssum[c], v);
    atomicAdd(&ssq[c], v * v);
  }
  __syncthreads();
  for (int i = threadIdx.x; i < C; i += blockDim.x) {
    atomicAdd(&stats[i], ssum[i]);
    atomicAdd(&stats[C + i], ssq[i]);
  }
}

// out = (res?res:0) + silu((x-mean)*rsqrt(var+eps)*g + be)
__global__ void bn_silu_kernel(const float* __restrict__ x, const float* __restrict__ stats,
                               const float* __restrict__ g, const float* __restrict__ be,
                               const float* __restrict__ res, float* out, long N, int C) {
  float invN = 1.0f / (float)N;
  long total = N * (long)C;
  long st = (long)gridDim.x * blockDim.x;
  for (long idx = (long)blockIdx.x * blockDim.x + threadIdx.x; idx < total; idx += st) {
    int c = (int)(idx % C);
    float mean = stats[c] * invN;
    float var = stats[C + c] * invN - mean * mean;
    float y = (x[idx] - mean) * rsqrtf(var + EPS_BN) * g[c] + be[c];
    float sil = y / (1.0f + expf(-y));
    out[idx] = (res ? res[idx] : 0.0f) + sil;
  }
}

// ---------------- softmax + entropy (one wave per row) ----------------

__global__ void softmax_ent_kernel(const float* __restrict__ x, float* s, float* ent_acc,
                                   long N, int K) {
  int wpb = blockDim.x >> 5;
  long row = (long)blockIdx.x * wpb + (threadIdx.x >> 5);
  if (row >= N) return;
  int lane = threadIdx.x & 31;
  const float* xr = x + row * K;
  float* sr = s + row * K;
  float mx = -3.402823466e38f;
  for (int c = lane; c < K; c += 32) mx = fmaxf(mx, xr[c]);
  for (int o = 16; o > 0; o >>= 1) mx = fmaxf(mx, __shfl_xor(mx, o));
  float sum = 0.f;
  for (int c = lane; c < K; c += 32) sum += expf(xr[c] - mx);
  for (int o = 16; o > 0; o >>= 1) sum += __shfl_xor(sum, o);
  float inv = 1.0f / sum;
  float ent = 0.f;
  for (int c = lane; c < K; c += 32) {
    float sv = expf(xr[c] - mx) * inv;
    sr[c] = sv;
    ent -= sv * logf(sv + EPS_ENT);
  }
  for (int o = 16; o > 0; o >>= 1) ent += __shfl_xor(ent, o);
  if (lane == 0) atomicAdd(ent_acc, ent);
}

// ---------------- level-2 dense helpers ----------------

__global__ void colsum_dinv_kernel(const float* __restrict__ apool, float* dinv2, int K) {
  int d = blockIdx.x * blockDim.x + threadIdx.x;
  if (d < K) {
    float sum = 0.f;
    for (int k = 0; k < K; k++) sum += apool[(long)k * K + d];
    sum += 1.0f;
    dinv2[d] = rsqrtf(fmaxf(sum, 1e-12f));
  }
}

// out[d,c] = dinv[d]*sum_k apool[k,d]*dinv[k]*h[k,c] + dinv[d]^2*h[d,c] + bias[c]
__global__ void dense_agg_kernel(const float* __restrict__ apool, const float* __restrict__ dinv,
                                 const float* __restrict__ h, const float* __restrict__ bias,
                                 float* out, int K, int C) {
  int d = blockIdx.x;
  int c = threadIdx.x;
  if (c >= C) return;
  float acc = 0.f;
  for (int k = 0; k < K; k++) acc += apool[(long)k * K + d] * dinv[k] * h[(long)k * C + c];
  float dd = dinv[d];
  out[(long)d * C + c] = dd * acc + dd * dd * h[(long)d * C + c] + bias[c];
}

__global__ void sumsq_kernel(const float* __restrict__ x, long n, float* out) {
  float acc = 0.f;
  long st = (long)gridDim.x * blockDim.x;
  for (long i = (long)blockIdx.x * blockDim.x + threadIdx.x; i < n; i += st) {
    float v = x[i];
    acc += v * v;
  }
  for (int o = 16; o > 0; o >>= 1) acc += __shfl_xor(acc, o);
  if ((threadIdx.x & 31) == 0) atomicAdd(out, acc);
}

// link = sqrt(max(ew2 - 2*trace(apool) + sum(sts^2), 0)) * invn2   (single block)
__global__ void link_kernel(const float* __restrict__ apool, const float* __restrict__ sts,
                            const float* __restrict__ ew2ptr, float ew2const,
                            int K, float invn2, float* out_link) {
  __shared__ float red[256];
  int tid = threadIdx.x;
  float tr = 0.f;
  for (int i = tid; i < K; i += blockDim.x) tr += apool[(long)i * K + i];
  float ss = 0.f;
  long kk = (long)K * K;
  for (long i = tid; i < kk; i += blockDim.x) { float v = sts[i]; ss += v * v; }
  red[tid] = tr; __syncthreads();
  for (int o = blockDim.x >> 1; o > 0; o >>= 1) { if (tid < o) red[tid] += red[tid + o]; __syncthreads(); }
  float TR = red[0]; __syncthreads();
  red[tid] = ss; __syncthreads();
  for (int o = blockDim.x >> 1; o > 0; o >>= 1) { if (tid < o) red[tid] += red[tid + o]; __syncthreads(); }
  if (tid == 0) {
    float SS = red[0];
    float ew2 = ew2ptr ? *ew2ptr : ew2const;
    out_link[0] = sqrtf(fmaxf(ew2 - 2.0f * TR + SS, 0.0f)) * invn2;
  }
}

// mean-pool rows -> linear -> log_softmax; out[OUTC] = aux loss  (single block, blockDim>=C)
__global__ void final_kernel(const float* __restrict__ xp2, const float* __restrict__ Wlin,
                             const float* __restrict__ blin,
                             const float* ent1, const float* ent2,
                             const float* link1, const float* link2,
                             float invN1, float invN2, int R, int C, int OUTC, float* out) {
  __shared__ float pooled[256];
  __shared__ float logits[64];
  int tid = threadIdx.x;
  if (tid < C) {
    float s = 0.f;
    for (int r = 0; r < R; r++) s += xp2[(long)r * C + tid];
    pooled[tid] = s / (float)R;
  }
  __syncthreads();
  if (tid < OUTC) {
    float a = blin[tid];
    for (int c = 0; c < C; c++) a += pooled[c] * Wlin[c * OUTC + tid];
    logits[tid] = a;
  }
  __syncthreads();
  if (tid == 0) {
    float mx = -3.402823466e38f;
    for (int o = 0; o < OUTC; o++) mx = fmaxf(mx, logits[o]);
    float se = 0.f;
    for (int o = 0; o < OUTC; o++) se += expf(logits[o] - mx);
    float lse = logf(se) + mx;
    for (int o = 0; o < OUTC; o++) out[o] = logits[o] - lse;
    out[OUTC] = ent1[0] * invN1 + ent2[0] * invN2 + link1[0] + link2[0];
  }
}

// ---------------- launcher ----------------

extern "C" void kernel_launch(void* const* d_in, const int* in_sizes, int n_in,
                              void* d_out, int out_size, void* d_ws, size_t ws_size,
                              hipStream_t stream) {
  const float* x0  = (const float*)d_in[0];
  const int*   eix = (const int*)d_in[1];
  const float* W1a = (const float*)d_in[4];  const float* b1a = (const float*)d_in[5];
  const float* g1a = (const float*)d_in[6];  const float* be1a= (const float*)d_in[7];
  const float* W1b = (const float*)d_in[8];  const float* b1b = (const float*)d_in[9];
  const float* g1b = (const float*)d_in[10]; const float* be1b= (const float*)d_in[11];
  const float* Wp1 = (const float*)d_in[12]; const float* bp1 = (const float*)d_in[13];
  const float* gp1 = (const float*)d_in[14]; const float* bep1= (const float*)d_in[15];
  const float* W2a = (const float*)d_in[16]; const float* b2a = (const float*)d_in[17];
  const float* g2a = (const float*)d_in[18]; const float* be2a= (const float*)d_in[19];
  const float* W2b = (const float*)d_in[20]; const float* b2b = (const float*)d_in[21];
  const float* g2b = (const float*)d_in[22]; const float* be2b= (const float*)d_in[23];
  const float* Wp2 = (const float*)d_in[24]; const float* bp2 = (const float*)d_in[25];
  const float* gp2 = (const float*)d_in[26]; const float* bep2= (const float*)d_in[27];
  const float* Wlin= (const float*)d_in[28]; const float* blin= (const float*)d_in[29];
  float* out = (float*)d_out;

  const int CH  = in_sizes[5];          // 128
  const int K1  = in_sizes[13];         // 256
  const int K2  = in_sizes[25];         // 64
  const int OUTC= in_sizes[29];         // 10
  const int IN  = in_sizes[4] / CH;     // 128
  const long N  = (long)in_sizes[0] / IN;  // 20000
  const int E   = in_sizes[1] / 2;      // 640000
  const int* src = eix;
  const int* dst = eix + E;

  // workspace carve-up (floats, 64-elt aligned)
  float* ws = (float*)d_ws;
  long off = 0;
  auto alloc = [&](long n) { float* p = ws + off; off += (n + 63) & ~63L; return p; };
  float* dinv1  = alloc(N);
  float* buf_h  = alloc(N * (long)K1);
  float* buf_agg= alloc(N * (long)K1);
  float* buf_xa = alloc(N * (long)CH);
  float* buf_xb = alloc(N * (long)CH);
  float* buf_s  = alloc(N * (long)K1);
  float* stats  = alloc(512);
  float* xp1    = alloc((long)K1 * CH);
  float* apool1 = alloc((long)K1 * K1);
  float* sts1   = alloc((long)K1 * K1);
  float* dinv2  = alloc(K1);
  float* h2     = alloc((long)K1 * CH);
  float* agg2   = alloc((long)K1 * CH);
  float* x2a    = alloc((long)K1 * CH);
  float* x2b    = alloc((long)K1 * CH);
  float* s2     = alloc((long)K1 * K2);
  float* xp2    = alloc((long)K2 * CH);
  float* t2     = alloc((long)K1 * K2);
  float* apool2 = alloc((long)K2 * K2);
  float* sts2   = alloc((long)K2 * K2);
  float* scal   = alloc(16);
  float* ent1 = scal + 0; float* ent2 = scal + 1;
  float* link1 = scal + 2; float* link2 = scal + 3;
  float* ew2sum2 = scal + 4;
  float* buf_t = buf_h;  // alias: free after pool GEMM+agg
  (void)ws_size; (void)n_in; (void)out_size;

  auto nb = [](long total, int bs) {
    long b = (total + bs - 1) / bs;
    if (b > 32768) b = 32768;
    if (b < 1) b = 1;
    return (int)b;
  };
  auto FILL = [&](float* p, float v, long n) {
    fill_f32<<<nb(n, 256), 256, 0, stream>>>(p, v, n);
  };
  auto GEMM_NN = [&](const float* A, const float* B, float* Cm, long M, int Nn, int Kc) {
    dim3 g((unsigned)(M / 16), (unsigned)(Nn / 16), 1);
    gemm_wmma_kernel<false, false><<<g, 32, 0, stream>>>(A, B, Cm, (int)M, Nn, Kc);
  };
  auto GEMM_TN_AT = [&](const float* A, const float* B, float* Cm, int M, int Nn, long Kc, int S) {
    dim3 g((unsigned)(M / 16), (unsigned)(Nn / 16), (unsigned)S);
    gemm_wmma_kernel<true, true><<<g, 32, 0, stream>>>(A, B, Cm, M, Nn, (int)Kc);
  };
  auto GEMM_TN = [&](const float* A, const float* B, float* Cm, int M, int Nn, long Kc) {
    dim3 g((unsigned)(M / 16), (unsigned)(Nn / 16), 1);
    gemm_wmma_kernel<true, false><<<g, 32, 0, stream>>>(A, B, Cm, M, Nn, (int)Kc);
  };
  auto AGG1 = [&](const float* h, const float* bias, float* o, int C) {
    selfloop_bias_kernel<<<nb(N * C, 256), 256, 0, stream>>>(h, dinv1, bias, o, N, C);
    edge_scatter_kernel<<<(E + 7) / 8, 256, 0, stream>>>(src, dst, dinv1, h, o, E, C);
  };
  auto BN_SILU = [&](const float* x, const float* g, const float* be, const float* res,
                     float* o, long rows, int C) {
    FILL(stats, 0.f, 2L * C);
    bn_stats_kernel<<<nb(rows * C, 256) < 512 ? nb(rows * C, 256) : 512, 256, 0, stream>>>(x, stats, rows, C);
    bn_silu_kernel<<<nb(rows * C, 256), 256, 0, stream>>>(x, stats, g, be, res, o, rows, C);
  };

  // ---- degree / normalization (level 1) ----
  FILL(dinv1, 0.f, N);
  deg_edges_kernel<<<(E + 255) / 256, 256, 0, stream>>>(dst, dinv1, E);
  dinv_kernel<<<nb(N, 256), 256, 0, stream>>>(dinv1, (int)N);

  // ---- level 1, block A ----
  GEMM_NN(x0, W1a, buf_h, N, CH, IN);
  AGG1(buf_h, b1a, buf_agg, CH);
  BN_SILU(buf_agg, g1a, be1a, nullptr, buf_xa, N, CH);

  // ---- level 1, block B (residual) ----
  GEMM_NN(buf_xa, W1b, buf_h, N, CH, CH);
  AGG1(buf_h, b1b, buf_agg, CH);
  BN_SILU(buf_agg, g1b, be1b, buf_xa, buf_xb, N, CH);

  // ---- level 1, pooling block ----
  GEMM_NN(buf_xb, Wp1, buf_h, N, K1, CH);
  AGG1(buf_h, bp1, buf_agg, K1);
  BN_SILU(buf_agg, gp1, bep1, nullptr, buf_agg, N, K1);
  FILL(ent1, 0.f, 1);
  softmax_ent_kernel<<<(int)((N + 7) / 8), 256, 0, stream>>>(buf_agg, buf_s, ent1, N, K1);

  // t[dst,k] += s[src,k]  (ew = 1); buf_t aliases buf_h (free now)
  FILL(buf_t, 0.f, N * (long)K1);
  edge_scatter_kernel<<<(E + 7) / 8, 256, 0, stream>>>(src, dst, nullptr, buf_s, buf_t, E, K1);

  // xp1 = s^T x ; apool1 = t^T s ; sts1 = s^T s   (split-K=50 over N=20000)
  FILL(xp1, 0.f, (long)K1 * CH);
  GEMM_TN_AT(buf_s, buf_xb, xp1, K1, CH, N, 50);
  FILL(apool1, 0.f, (long)K1 * K1);
  GEMM_TN_AT(buf_t, buf_s, apool1, K1, K1, N, 50);
  FILL(sts1, 0.f, (long)K1 * K1);
  GEMM_TN_AT(buf_s, buf_s, sts1, K1, K1, N, 50);
  link_kernel<<<1, 256, 0, stream>>>(apool1, sts1, nullptr, (float)E, K1,
                                     1.0f / ((float)N * (float)N), link1);

  // ---- level 2 setup ----
  colsum_dinv_kernel<<<(K1 + 255) / 256, 256, 0, stream>>>(apool1, dinv2, K1);
  FILL(ew2sum2, 0.f, 1);
  sumsq_kernel<<<64, 256, 0, stream>>>(apool1, (long)K1 * K1, ew2sum2);

  // ---- level 2, block A ----
  GEMM_NN(xp1, W2a, h2, K1, CH, CH);
  dense_agg_kernel<<<K1, CH, 0, stream>>>(apool1, dinv2, h2, b2a, agg2, K1, CH);
  BN_SILU(agg2, g2a, be2a, nullptr, x2a, K1, CH);

  // ---- level 2, block B (residual) ----
  GEMM_NN(x2a, W2b, h2, K1, CH, CH);
  dense_agg_kernel<<<K1, CH, 0, stream>>>(apool1, dinv2, h2, b2b, agg2, K1, CH);
  BN_SILU(agg2, g2b, be2b, x2a, x2b, K1, CH);

  // ---- level 2, pooling block ----
  GEMM_NN(x2b, Wp2, h2, K1, K2, CH);
  dense_agg_kernel<<<K1, K2, 0, stream>>>(apool1, dinv2, h2, bp2, agg2, K1, K2);
  BN_SILU(agg2, gp2, bep2, nullptr, agg2, K1, K2);
  FILL(ent2, 0.f, 1);
  softmax_ent_kernel<<<(K1 + 7) / 8, 256, 0, stream>>>(agg2, s2, ent2, K1, K2);

  // xp2 = s2^T x2b ; t2 = apool1^T s2 ; apool2 = t2^T s2 ; sts2 = s2^T s2
  GEMM_TN(s2, x2b, xp2, K2, CH, K1);
  GEMM_TN(apool1, s2, t2, K1, K2, K1);
  GEMM_TN(t2, s2, apool2, K2, K2, K1);
  GEMM_TN(s2, s2, sts2, K2, K2, K1);
  link_kernel<<<1, 256, 0, stream>>>(apool2, sts2, ew2sum2, 0.f, K2,
                                     1.0f / ((float)K1 * (float)K1), link2);

  // ---- head: mean pool over K2 rows -> linear -> log_softmax ; aux losses ----
  final_kernel<<<1, 128, 0, stream>>>(xp2, Wlin, blin, ent1, ent2, link1, link2,
                                      1.0f / (float)N, 1.0f / (float)K1, K2, CH, OUTC, out);
}